// MambaBlock_61753039782195
// MI455X (gfx1250) — compile-verified
//
#include <hip/hip_runtime.h>
#include <math.h>

// ---------------------------------------------------------------------------
// Problem constants (match reference)
// ---------------------------------------------------------------------------
#define BB      2
#define LL      2048
#define DMODEL  1024
#define DINNER  2048
#define DSTATE  16
#define DCONV   4
#define DTRANK  64
#define HMLP    4096
#define NTOK    (BB * LL)       // 4096
#define EPSV    1e-5f

typedef __attribute__((ext_vector_type(2))) float v2f;
typedef __attribute__((ext_vector_type(8))) float v8f;
typedef int v4i_vs __attribute__((vector_size(16)));   // matches builtin pointee

enum { EP_NONE = 0, EP_BIAS_SOFTPLUS = 1, EP_RESIDUAL = 2, EP_GLU = 3 };

__device__ __forceinline__ float silu_f(float x) { return x / (1.0f + expf(-x)); }
__device__ __forceinline__ float softplus_f(float x) {
    return (x > 20.0f) ? x : log1pf(expf(x));
}

// ---------------------------------------------------------------------------
// CDNA5 async global->LDS copy (ASYNCcnt-tracked), guarded by __has_builtin
// so the file compiles on toolchains without the builtin (falls back to the
// synchronous reg-staged path).
// Builtin signature (from hipcc diagnostic): 
//   void __builtin_amdgcn_global_load_async_to_lds_b128(
//        v4i addrspace(1)* gsrc, v4i addrspace(3)* ldst, Ii offset, Ii cpol)
// ---------------------------------------------------------------------------
#if defined(__has_builtin)
#  if __has_builtin(__builtin_amdgcn_global_load_async_to_lds_b128)
#    define USE_ASYNC_LDS 1
#  else
#    define USE_ASYNC_LDS 0
#  endif
#else
#  define USE_ASYNC_LDS 0
#endif

#if USE_ASYNC_LDS
#define ASYNC_CP_B128(gsrc, ldst)                                              \
    __builtin_amdgcn_global_load_async_to_lds_b128(                            \
        (__attribute__((address_space(1))) v4i_vs*)(gsrc),                     \
        (__attribute__((address_space(3))) v4i_vs*)(ldst), 0, 0)
#if __has_builtin(__builtin_amdgcn_s_wait_asynccnt)
#define WAIT_ASYNC() __builtin_amdgcn_s_wait_asynccnt(0)
#else
#define WAIT_ASYNC() asm volatile("s_wait_asynccnt 0x0" ::: "memory")
#endif
#endif

// ---------------------------------------------------------------------------
// f32 WMMA GEMM with LDS-staged, double-buffered tiles.
//   C[M,N] = A[M,K] @ B[K,N] (+ epilogue)
// Block: 256 threads = 8 waves (4 along M, 2 along N). Block tile 128x64.
// K consumed in chunks of 32 (8 k-steps of v_wmma_f32_16x16x4_f32).
// Non-BOUND path: chunks are copied global->LDS with async B128 copies
// (no VGPR round-trip, one barrier per chunk, ASYNCcnt-synchronized).
// BOUND path (N=96 GEMM only): synchronous guarded loads.
// EP_GLU additionally stages the gate chunk (cols +gate_off) and stores
// val*silu(gate)  (fused SwiGLU for fc1).
// ---------------------------------------------------------------------------
constexpr int MBT = 128;   // block tile M
constexpr int NBT = 64;    // block tile N
constexpr int KC  = 32;    // K chunk
constexpr int KCP = 36;    // padded A row stride: 144B = 9*16 (B128-aligned)
constexpr int NBP = 68;    // padded B row stride: 272B = 17*16 (B128-aligned)

template <int EP, bool BOUND>
__global__ __launch_bounds__(256)
void wmma_gemm_f32_kernel(const float* __restrict__ A, int lda,
                          const float* __restrict__ B, int ldb,
                          float*       __restrict__ C, int ldc,
                          int N, int K,
                          const float* __restrict__ aux, int aux_ld,
                          int gate_off)
{
    __shared__ float sA[2][MBT * KCP];
    __shared__ float sB[2][(EP == EP_GLU ? 2 : 1) * KC * NBP];

    const int tid  = threadIdx.x;
    const int lane = tid & 31;
    const int wave = tid >> 5;
    const int half = lane >> 4;    // 0: K{0,1} | 1: K{2,3} of each k-step
    const int lr   = lane & 15;

    const int m_blk = blockIdx.y * MBT;
    const int n_blk = blockIdx.x * NBT;
    const int m_loc = (wave & 3) * 32;   // wave origin inside block tile
    const int n_loc = (wave >> 2) * 32;

    const int nchunks = K / KC;

    v8f acc[2][2]  = {};
    v8f gacc[2][2] = {};

    // 8 branch-free wmma k-steps out of LDS buffer `buf`.
    auto compute = [&](int buf) {
#pragma unroll
        for (int ks = 0; ks < 8; ++ks) {
            const int kk = ks * 4 + 2 * half;
            v2f af[2], bf[2], gf[2];
#pragma unroll
            for (int mi = 0; mi < 2; ++mi)
                af[mi] = *(const v2f*)&sA[buf][(m_loc + 16 * mi + lr) * KCP + kk];
#pragma unroll
            for (int ni = 0; ni < 2; ++ni) {
                const int nn = n_loc + 16 * ni + lr;
                v2f b;
                b.x = sB[buf][kk * NBP + nn];
                b.y = sB[buf][(kk + 1) * NBP + nn];
                bf[ni] = b;
                if (EP == EP_GLU) {
                    v2f g;
                    g.x = sB[buf][KC * NBP + kk * NBP + nn];
                    g.y = sB[buf][KC * NBP + (kk + 1) * NBP + nn];
                    gf[ni] = g;
                }
            }
#pragma unroll
            for (int mi = 0; mi < 2; ++mi)
#pragma unroll
                for (int ni = 0; ni < 2; ++ni) {
                    acc[mi][ni] = __builtin_amdgcn_wmma_f32_16x16x4_f32(
                        false, af[mi], false, bf[ni],
                        (short)0, acc[mi][ni], false, false);
                    if (EP == EP_GLU)
                        gacc[mi][ni] = __builtin_amdgcn_wmma_f32_16x16x4_f32(
                            false, af[mi], false, gf[ni],
                            (short)0, gacc[mi][ni], false, false);
                }
        }
    };

#if USE_ASYNC_LDS
    if constexpr (!BOUND) {
        // --- Async double-buffered pipeline (CDNA5 ASYNCcnt path) ---
        // A chunk: 128x32 f32 = 1024 B128 slots, 4 per thread.
        // B chunk: 32x64  f32 =  512 B128 slots, 2 per thread (+2 for gate).
        auto issue = [&](int c, int buf) {
#pragma unroll
            for (int i = 0; i < 4; ++i) {
                const int slot = tid + 256 * i;
                const int row  = slot >> 3;
                const int kq   = (slot & 7) << 2;
                ASYNC_CP_B128(A + (size_t)(m_blk + row) * lda + c * KC + kq,
                              &sA[buf][row * KCP + kq]);
            }
#pragma unroll
            for (int i = 0; i < 2; ++i) {
                const int idx = tid + 256 * i;
                const int kr  = idx >> 4;
                const int nc  = (idx & 15) << 2;
                const float* bp = B + (size_t)(c * KC + kr) * ldb + n_blk + nc;
                ASYNC_CP_B128(bp, &sB[buf][kr * NBP + nc]);
                if (EP == EP_GLU)
                    ASYNC_CP_B128(bp + gate_off, &sB[buf][KC * NBP + kr * NBP + nc]);
            }
        };

        issue(0, 0);
        WAIT_ASYNC();
        __syncthreads();

        for (int c = 0; c < nchunks; ++c) {
            if (c + 1 < nchunks) issue(c + 1, (c + 1) & 1);
            compute(c & 1);
            if (c + 1 < nchunks) WAIT_ASYNC();
            __syncthreads();
        }
    } else
#endif
    {
        // --- Synchronous reg-staged pipeline (bounds-checked / fallback) ---
        float4 pa[4];
        float  pb[8], pg[8];

        auto gload = [&](int c) {
#pragma unroll
            for (int i = 0; i < 4; ++i) {
                const int slot = tid + 256 * i;
                const int row  = slot >> 3;
                const int kq   = (slot & 7) << 2;
                pa[i] = *(const float4*)(A + (size_t)(m_blk + row) * lda + c * KC + kq);
            }
#pragma unroll
            for (int i = 0; i < 8; ++i) {
                const int idx = tid + 256 * i;
                const int kr  = idx >> 6;
                const int nc  = idx & 63;
                const float* bp = B + (size_t)(c * KC + kr) * ldb + n_blk + nc;
                const bool ok = !BOUND || (n_blk + nc) < N;
                pb[i] = ok ? bp[0] : 0.0f;
                if (EP == EP_GLU) pg[i] = ok ? bp[gate_off] : 0.0f;
            }
            if (c + 1 < nchunks) {
                __builtin_prefetch(A + (size_t)(m_blk + (tid >> 1)) * lda + (c + 1) * KC, 0, 0);
                __builtin_prefetch(B + (size_t)((c + 1) * KC + (tid >> 3)) * ldb + n_blk, 0, 0);
            }
        };
        auto sstore = [&](int buf) {
#pragma unroll
            for (int i = 0; i < 4; ++i) {
                const int slot = tid + 256 * i;
                const int row  = slot >> 3;
                const int kq   = (slot & 7) << 2;
                *(float4*)&sA[buf][row * KCP + kq] = pa[i];
            }
#pragma unroll
            for (int i = 0; i < 8; ++i) {
                const int idx = tid + 256 * i;
                const int kr  = idx >> 6;
                const int nc  = idx & 63;
                sB[buf][kr * NBP + nc] = pb[i];
                if (EP == EP_GLU) sB[buf][KC * NBP + kr * NBP + nc] = pg[i];
            }
        };

        gload(0);
        sstore(0);
        __syncthreads();

        for (int c = 0; c < nchunks; ++c) {
            if (c + 1 < nchunks) gload(c + 1);
            compute(c & 1);
            __syncthreads();
            if (c + 1 < nchunks) {
                sstore((c + 1) & 1);
                __syncthreads();
            }
        }
    }

    // Store: C/D layout — VGPR v holds M = v (lanes 0-15) / v+8 (lanes 16-31)
#pragma unroll
    for (int mi = 0; mi < 2; ++mi)
#pragma unroll
        for (int ni = 0; ni < 2; ++ni) {
            const int n = n_blk + n_loc + 16 * ni + lr;
            if (BOUND && n >= N) continue;
#pragma unroll
            for (int v = 0; v < 8; ++v) {
                const int m = m_blk + m_loc + 16 * mi + v + 8 * half;
                float val = acc[mi][ni][v];
                if (EP == EP_BIAS_SOFTPLUS) {
                    val = softplus_f(val + aux[n]);
                } else if (EP == EP_RESIDUAL) {
                    val += aux[(size_t)m * aux_ld + n];
                } else if (EP == EP_GLU) {
                    val *= silu_f(gacc[mi][ni][v]);
                }
                C[(size_t)m * ldc + n] = val;
            }
        }
}

// ---------------------------------------------------------------------------
// Depthwise causal conv (k=4) + bias + SiLU.
// xz: [NTOK, 2*DINNER] (first DINNER cols = xc path) -> xc: [NTOK, DINNER]
// ---------------------------------------------------------------------------
__global__ __launch_bounds__(256)
void conv_silu_kernel(const float* __restrict__ xz,
                      const float* __restrict__ conv_w,
                      const float* __restrict__ conv_b,
                      float* __restrict__ xc)
{
    const size_t idx = (size_t)blockIdx.x * blockDim.x + threadIdx.x;
    if (idx >= (size_t)NTOK * DINNER) return;
    const int d = (int)(idx % DINNER);
    const int t = (int)((idx / DINNER) % LL);
    const int b = (int)(idx / ((size_t)DINNER * LL));

    float acc = conv_b[d];
#pragma unroll
    for (int k = 0; k < DCONV; ++k) {
        const int tt = t - (DCONV - 1) + k;
        if (tt >= 0)
            acc += xz[((size_t)b * LL + tt) * (2 * DINNER) + d] * conv_w[d * DCONV + k];
    }
    xc[idx] = silu_f(acc);
}

// ---------------------------------------------------------------------------
// Selective scan. One thread per (b, d) channel; 16 states in registers.
// B/C (shared per token) staged through LDS by the first wave.
// Fuses: y = scan + D*xc, then y *= silu(z).
// ---------------------------------------------------------------------------
__global__ __launch_bounds__(256)
void scan_kernel(const float* __restrict__ dt,
                 const float* __restrict__ xc,
                 const float* __restrict__ dbl,   // [NTOK,96]: [0:64)=dt, [64:80)=B, [80:96)=C
                 const float* __restrict__ xz,    // for z (cols [DINNER, 2*DINNER))
                 const float* __restrict__ A_log,
                 const float* __restrict__ Dp,
                 float* __restrict__ y)
{
    __shared__ float sB[DSTATE];
    __shared__ float sC[DSTATE];

    const int b = blockIdx.x >> 3;                       // 2 batches x 8 blocks
    const int d = ((blockIdx.x & 7) << 8) + threadIdx.x; // 256 channels / block

    float Ad[DSTATE], h[DSTATE];
#pragma unroll
    for (int s = 0; s < DSTATE; ++s) {
        Ad[s] = -expf(A_log[d * DSTATE + s]);
        h[s]  = 0.0f;
    }
    const float Dd = Dp[d];

    for (int t = 0; t < LL; ++t) {
        const size_t tok = (size_t)b * LL + t;
        if (threadIdx.x < 2 * DSTATE) {
            const int s = threadIdx.x & (DSTATE - 1);
            const float v = dbl[tok * 96 + DTRANK + (threadIdx.x >> 4) * DSTATE + s];
            if (threadIdx.x < DSTATE) sB[s] = v; else sC[s] = v;
        }
        __syncthreads();

        const float dtv = dt[tok * DINNER + d];
        const float xv  = xc[tok * DINNER + d];
        const float dtx = dtv * xv;
        float ys = 0.0f;
#pragma unroll
        for (int s = 0; s < DSTATE; ++s) {
            h[s] = expf(dtv * Ad[s]) * h[s] + dtx * sB[s];
            ys += h[s] * sC[s];
        }
        float yv = ys + Dd * xv;
        const float zv = xz[tok * (2 * DINNER) + DINNER + d];
        yv *= silu_f(zv);
        y[tok * DINNER + d] = yv;
        __syncthreads();
    }
}

// ---------------------------------------------------------------------------
// RMSNorm over last dim (1024), one block (256 thr) per row.
// ---------------------------------------------------------------------------
__global__ __launch_bounds__(256)
void rmsnorm_kernel(const float* __restrict__ x,
                    const float* __restrict__ w,
                    float* __restrict__ out)
{
    __shared__ float red[8];
    const int row = blockIdx.x;
    const float* xr = x + (size_t)row * DMODEL;

    float ss = 0.0f;
    for (int i = threadIdx.x; i < DMODEL; i += 256) {
        const float v = xr[i];
        ss += v * v;
    }
#pragma unroll
    for (int off = 16; off > 0; off >>= 1)
        ss += __shfl_down(ss, off, 32);
    if ((threadIdx.x & 31) == 0) red[threadIdx.x >> 5] = ss;
    __syncthreads();
    if (threadIdx.x == 0) {
        float t = 0.0f;
        for (int i = 0; i < 8; ++i) t += red[i];
        red[0] = rsqrtf(t / (float)DMODEL + EPSV);
    }
    __syncthreads();
    const float sc = red[0];
    for (int i = threadIdx.x; i < DMODEL; i += 256)
        out[(size_t)row * DMODEL + i] = xr[i] * sc * w[i];
}

// ---------------------------------------------------------------------------
// Launcher
// ---------------------------------------------------------------------------
extern "C" void kernel_launch(void* const* d_in, const int* in_sizes, int n_in,
                              void* d_out, int out_size, void* d_ws, size_t ws_size,
                              hipStream_t stream)
{
    const float* x       = (const float*)d_in[0];
    const float* W_in    = (const float*)d_in[1];
    const float* conv_w  = (const float*)d_in[2];
    const float* conv_b  = (const float*)d_in[3];
    const float* W_x     = (const float*)d_in[4];
    const float* W_dt    = (const float*)d_in[5];
    const float* dt_bias = (const float*)d_in[6];
    const float* A_log   = (const float*)d_in[7];
    const float* Dp      = (const float*)d_in[8];
    const float* W_out   = (const float*)d_in[9];
    const float* rn_w    = (const float*)d_in[10];
    const float* fc1_w   = (const float*)d_in[11];
    const float* fc2_w   = (const float*)d_in[12];
    float* out = (float*)d_out;

    char* ws = (char*)d_ws;
    const size_t MB = 1024ull * 1024ull;
    float* xz  = (float*)(ws + 0);           // 4096x4096 f32 = 64 MB
    float* xc  = (float*)(ws + 64 * MB);     // 4096x2048      = 32 MB
    float* dtb = (float*)(ws + 96 * MB);     // 4096x2048      = 32 MB
    float* yb  = (float*)(ws + 128 * MB);    // 4096x2048      = 32 MB
    float* x1  = (float*)(ws + 160 * MB);    // 4096x1024      = 16 MB
    float* xn  = (float*)(ws + 176 * MB);    // 4096x1024      = 16 MB
    float* dbl = (float*)(ws + 192 * MB);    // 4096x96        = 1.5 MB
    float* hb  = xz;                         // reuse xz region for MLP hidden (64 MB)

    dim3 blk(256);

    // 1) xz = x @ W_in                    M=4096 K=1024 N=4096
    wmma_gemm_f32_kernel<EP_NONE, false><<<dim3(4096 / 64, NTOK / 128), blk, 0, stream>>>(
        x, DMODEL, W_in, 2 * DINNER, xz, 2 * DINNER, 2 * DINNER, DMODEL, nullptr, 0, 0);

    // 2) depthwise causal conv + SiLU -> xc
    conv_silu_kernel<<<(NTOK * DINNER) / 256, blk, 0, stream>>>(xz, conv_w, conv_b, xc);

    // 3) dbl = xc @ W_x                  M=4096 K=2048 N=96 (bounds-checked)
    wmma_gemm_f32_kernel<EP_NONE, true><<<dim3(2, NTOK / 128), blk, 0, stream>>>(
        xc, DINNER, W_x, DTRANK + 2 * DSTATE, dbl, DTRANK + 2 * DSTATE,
        DTRANK + 2 * DSTATE, DINNER, nullptr, 0, 0);

    // 4) dt = softplus(dbl[:, :64] @ W_dt + dt_bias)   M=4096 K=64 N=2048
    wmma_gemm_f32_kernel<EP_BIAS_SOFTPLUS, false><<<dim3(2048 / 64, NTOK / 128), blk, 0, stream>>>(
        dbl, DTRANK + 2 * DSTATE, W_dt, DINNER, dtb, DINNER, DINNER, DTRANK,
        dt_bias, 0, 0);

    // 5) selective scan (fuses +D*xc and *silu(z))
    scan_kernel<<<BB * (DINNER / 256), blk, 0, stream>>>(dtb, xc, dbl, xz, A_log, Dp, yb);

    // 6) x1 = y @ W_out + x              M=4096 K=2048 N=1024
    wmma_gemm_f32_kernel<EP_RESIDUAL, false><<<dim3(1024 / 64, NTOK / 128), blk, 0, stream>>>(
        yb, DINNER, W_out, DMODEL, x1, DMODEL, DMODEL, DINNER, x, DMODEL, 0);

    // 7) xn = rmsnorm(x1)
    rmsnorm_kernel<<<NTOK, blk, 0, stream>>>(x1, rn_w, xn);

    // 8) h = (xn@fc1)[:, :4096] * silu((xn@fc1)[:, 4096:])  (fused GLU)
    wmma_gemm_f32_kernel<EP_GLU, false><<<dim3(HMLP / 64, NTOK / 128), blk, 0, stream>>>(
        xn, DMODEL, fc1_w, 2 * HMLP, hb, HMLP, HMLP, DMODEL, nullptr, 0, HMLP);

    // 9) out = h @ fc2 + x1              M=4096 K=4096 N=1024
    wmma_gemm_f32_kernel<EP_RESIDUAL, false><<<dim3(1024 / 64, NTOK / 128), blk, 0, stream>>>(
        hb, HMLP, fc2_w, DMODEL, out, DMODEL, DMODEL, HMLP, x1, DMODEL, 0);
}